// MambaBlock_4913442586861
// MI455X (gfx1250) — compile-verified
//
#include <hip/hip_runtime.h>
#include <hip/hip_bf16.h>
#include <math.h>

// ---------------- problem constants ----------------
#define BATCH   2
#define SEQLEN  2048
#define DMODEL  1024
#define DINNER  2048
#define DSTATE  16
#define DTRANK  64
#define DCONV   4
#define MTOT    (BATCH * SEQLEN)   // 4096 rows for all GEMMs

typedef float v2f __attribute__((ext_vector_type(2)));
typedef float v8f __attribute__((ext_vector_type(8)));

// ---- Tensor Data Mover availability (clang-23 / therock 6-arg form) ----
#if defined(__HIP_DEVICE_COMPILE__)
#if __has_builtin(__builtin_amdgcn_tensor_load_to_lds) && __has_builtin(__builtin_amdgcn_s_wait_tensorcnt)
#define USE_TDM 1
#endif
#endif
#ifndef USE_TDM
#define USE_TDM 0
#endif

#if USE_TDM
typedef unsigned int v4u __attribute__((ext_vector_type(4)));
typedef int          v4i __attribute__((ext_vector_type(4)));
typedef int          v8i __attribute__((ext_vector_type(8)));

__device__ inline unsigned lds_off(const void* p) {
    // generic LDS pointer: addr[31:0] == workgroup-relative LDS byte offset
    return (unsigned)(unsigned long long)(size_t)p;
}

// 2D TDM tile load: tile = 128 rows x 32 floats, row stride `stride` floats,
// LDS padded +4 DWORDs every 32 DWORDs -> 36-float LDS rows.
// remRows < 128 => OOB rows read as zero (ISA: reads beyond tensor dim return 0).
__device__ inline void tdm_load_2d(unsigned ldsByteOff, const float* gsrc,
                                   int remK, int remRows, int stride) {
    unsigned long long ga = (unsigned long long)(size_t)gsrc;
    v4u g0;
    g0[0] = 1u;                                                   // count=1, no gather
    g0[1] = ldsByteOff;                                           // lds_addr
    g0[2] = (unsigned)ga;                                         // global_addr[31:0]
    g0[3] = (unsigned)((ga >> 32) & 0x01ffffffull) | (2u << 30);  // addr[56:32] | type=2
    v8i g1;
    // data_size=4B (2), pad_enable, pad_interval=32 DW (enc 4), pad_amount=4 DW (enc 3)
    g1[0] = (int)((2u << 16) | (1u << 20) | (4u << 22) | (3u << 25));
    g1[1] = (int)(((unsigned)remK & 0xffffu) << 16);              // tensor_dim0[15:0]
    g1[2] = (int)((((unsigned)remK >> 16) & 0xffffu)              // tensor_dim0[31:16]
                  | (((unsigned)remRows & 0xffffu) << 16));       // tensor_dim1[15:0]
    g1[3] = (int)((((unsigned)remRows >> 16) & 0xffffu)           // tensor_dim1[31:16]
                  | (32u << 16));                                 // tile_dim0 = 32
    g1[4] = 128;                                                  // tile_dim1=128, tile_dim2=0
    g1[5] = stride;                                               // tensor_dim0_stride[31:0]
    g1[6] = 0;
    g1[7] = 0;
    v4i z4 = {0, 0, 0, 0};
    v8i z8 = {0, 0, 0, 0, 0, 0, 0, 0};
    __builtin_amdgcn_tensor_load_to_lds(g0, g1, z4, z4, z8, 0);
}
#endif  // USE_TDM

// =====================================================================
// Fp32 WMMA TN GEMM:  C[M,N] = A[M,K] * B[N,K]^T  (both K-contiguous)
// Block tile 128x128, K-chunk 32, 8 waves: wave tile 64(M) x 32(N),
// V_WMMA_F32_16X16X4_F32 (exact fp32). TDM double-buffered LDS pipeline.
// EPI: 0 = none, 1 = softplus(acc + bias[col])
// =====================================================================
template <int EPI>
__global__ __launch_bounds__(256)
void gemm_tn_wmma(const float* __restrict__ A, int lda,
                  const float* __restrict__ Bw, int ldb,
                  float* __restrict__ C, int ldc,
                  int M, int N, int K,
                  const float* __restrict__ bias) {
    const int tid  = threadIdx.x;
    const int lane = tid & 31;
    const int wave = tid >> 5;
    const int l16  = lane & 15;
    const int half = lane >> 4;         // 0: K={k,k+1}, 1: K={k+2,k+3}
    const int wmOff = (wave & 1) * 64;  // wave M offset in block tile
    const int wnOff = (wave >> 1) * 32; // wave N offset in block tile

    const int m0 = blockIdx.y * 128;
    const int n0 = blockIdx.x * 128;

    v8f acc[4][2];
#pragma unroll
    for (int mi = 0; mi < 4; ++mi)
#pragma unroll
        for (int ni = 0; ni < 2; ++ni)
            acc[mi][ni] = v8f{0.f, 0.f, 0.f, 0.f, 0.f, 0.f, 0.f, 0.f};

    // 64 WMMAs on one staged K-chunk
    auto compute_chunk = [&](const float (*pA)[36], const float (*pB)[36]) {
#pragma unroll
        for (int kk = 0; kk < 32; kk += 4) {
            const int ko = kk + half * 2;
            v2f af[4], bf[2];
#pragma unroll
            for (int mi = 0; mi < 4; ++mi)
                af[mi] = *(const v2f*)&pA[wmOff + mi * 16 + l16][ko];
#pragma unroll
            for (int ni = 0; ni < 2; ++ni)
                bf[ni] = *(const v2f*)&pB[wnOff + ni * 16 + l16][ko];
#pragma unroll
            for (int mi = 0; mi < 4; ++mi)
#pragma unroll
                for (int ni = 0; ni < 2; ++ni)
                    acc[mi][ni] = __builtin_amdgcn_wmma_f32_16x16x4_f32(
                        /*neg_a=*/false, af[mi], /*neg_b=*/false, bf[ni],
                        /*c_mod=*/(short)0, acc[mi][ni],
                        /*reuse_a=*/false, /*reuse_b=*/false);
        }
    };

#if USE_TDM
    // ---- TDM double-buffered pipeline: DMA tiles, lanes only do WMMA ----
    __shared__ __align__(16) float sA[2][128][36];
    __shared__ __align__(16) float sB[2][128][36];

    const int nCh     = K >> 5;
    const int rowsA   = M - m0;          // >= 128, always in-bounds
    const int rowsB   = N - n0;          // may be < 128 -> TDM zero-fills OOB rows
    if (wave == 0) {
        tdm_load_2d(lds_off(&sA[0][0][0]), A  + (size_t)m0 * lda, K, rowsA, lda);
        tdm_load_2d(lds_off(&sB[0][0][0]), Bw + (size_t)n0 * ldb, K, rowsB, ldb);
    }
    for (int ic = 0; ic < nCh; ++ic) {
        const int cur = ic & 1;
        if (wave == 0) __builtin_amdgcn_s_wait_tensorcnt(0);  // buf[cur] resident
        __syncthreads();            // gate all waves; also: buf[cur^1] reads done
        if (wave == 0 && ic + 1 < nCh) {
            const int kc2 = (ic + 1) << 5;
            tdm_load_2d(lds_off(&sA[cur ^ 1][0][0]),
                        A  + (size_t)m0 * lda + kc2, K - kc2, rowsA, lda);
            tdm_load_2d(lds_off(&sB[cur ^ 1][0][0]),
                        Bw + (size_t)n0 * ldb + kc2, K - kc2, rowsB, ldb);
        }
        compute_chunk(sA[cur], sB[cur]);
    }
#else
    // ---- fallback: synchronous float4 staging (known-good path) ----
    (void)M;
    __shared__ __align__(16) float sA1[128][36];
    __shared__ __align__(16) float sB1[128][36];
    for (int kc = 0; kc < K; kc += 32) {
        __syncthreads();
#pragma unroll
        for (int i = 0; i < 4; ++i) {
            int slot = tid + i * 256;
            int row  = slot >> 3;
            int kg   = (slot & 7) << 2;
            float4 av = *(const float4*)&A[(size_t)(m0 + row) * lda + kc + kg];
            *(float4*)&sA1[row][kg] = av;
            float4 bv = make_float4(0.f, 0.f, 0.f, 0.f);
            int n = n0 + row;
            if (n < N) bv = *(const float4*)&Bw[(size_t)n * ldb + kc + kg];
            *(float4*)&sB1[row][kg] = bv;
        }
        __syncthreads();
        compute_chunk(sA1, sB1);
    }
#endif

    // ---- epilogue / store: VGPR r -> row M = r + half*8 ----
#pragma unroll
    for (int mi = 0; mi < 4; ++mi)
#pragma unroll
        for (int ni = 0; ni < 2; ++ni)
#pragma unroll
            for (int r = 0; r < 8; ++r) {
                int row = m0 + wmOff + mi * 16 + half * 8 + r;
                int col = n0 + wnOff + ni * 16 + l16;
                if (col < N) {
                    float v = acc[mi][ni][r];
                    if (EPI == 1) {
                        v += bias[col];
                        v = (v > 20.f) ? v : log1pf(__expf(v));  // softplus
                    }
                    C[(size_t)row * ldc + col] = v;
                }
            }
}

// =====================================================================
// Depthwise causal conv1d (width 4) + bias + SiLU, on xc = xz[:, :, :2048]
// =====================================================================
__global__ __launch_bounds__(256)
void conv_silu_kernel(const float* __restrict__ xz,
                      const float* __restrict__ conv_w,
                      const float* __restrict__ conv_b,
                      float* __restrict__ xconv) {
    size_t idx = (size_t)blockIdx.x * blockDim.x + threadIdx.x;  // B*L*ED = 2^23
    int e = idx & (DINNER - 1);
    int l = (idx >> 11) & (SEQLEN - 1);
    int b = (int)(idx >> 22);
    const float* base = xz + (size_t)b * SEQLEN * (2 * DINNER);
    float acc = conv_b[e];
#pragma unroll
    for (int k = 0; k < DCONV; ++k) {
        int ls = l + k - (DCONV - 1);
        if (ls >= 0) acc += conv_w[e * DCONV + k] * base[(size_t)ls * (2 * DINNER) + e];
    }
    float s = acc / (1.f + __expf(-acc));  // silu
    xconv[idx] = s;
}

// =====================================================================
// Selective scan: lane = (e, n) pair (wave32: 2 channels x 16 states).
// h_l = exp(delta*A)*h_{l-1} + delta*B_l*x_l ; y = sum_n h*C + D*x ;
// fused with y * silu(z) so out_proj is a plain GEMM afterwards.
// grid = BATCH * (DINNER/16) blocks of 256 threads (8 waves = 16 channels)
// =====================================================================
__global__ __launch_bounds__(256)
void scan_kernel(const float* __restrict__ xconv,
                 const float* __restrict__ dbc,
                 const float* __restrict__ delta,
                 const float* __restrict__ A_log,
                 const float* __restrict__ D_param,
                 const float* __restrict__ xz,
                 float* __restrict__ ycomb) {
    const int tid  = threadIdx.x;
    const int lane = tid & 31;
    const int wave = tid >> 5;
    const int blk  = blockIdx.x;                  // 0..255
    const int b    = blk / (DINNER / 16);
    const int eg   = blk % (DINNER / 16);
    const int e    = eg * 16 + wave * 2 + (lane >> 4);
    const int n    = lane & 15;

    const float Aen = -__expf(A_log[e * DSTATE + n]);
    const float Dv  = D_param[e];
    float h = 0.f;

    const size_t rowBase = (size_t)b * SEQLEN;
    for (int l = 0; l < SEQLEN; ++l) {
        const size_t row = rowBase + l;
        float dlt = delta[row * DINNER + e];
        float xv  = xconv[row * DINNER + e];
        float Bn  = dbc[row * (DTRANK + 2 * DSTATE) + DTRANK + n];
        float Cn  = dbc[row * (DTRANK + 2 * DSTATE) + DTRANK + DSTATE + n];
        h = __expf(dlt * Aen) * h + dlt * Bn * xv;
        float p = h * Cn;
        p += __shfl_xor(p, 1);
        p += __shfl_xor(p, 2);
        p += __shfl_xor(p, 4);
        p += __shfl_xor(p, 8);            // reduce over 16 states (stays in half-wave)
        if (n == 0) {
            float y  = p + Dv * xv;
            float zv = xz[row * (2 * DINNER) + DINNER + e];
            float sz = zv / (1.f + __expf(-zv));
            ycomb[row * DINNER + e] = y * sz;
        }
    }
}

// =====================================================================
extern "C" void kernel_launch(void* const* d_in, const int* in_sizes, int n_in,
                              void* d_out, int out_size, void* d_ws, size_t ws_size,
                              hipStream_t stream) {
    (void)in_sizes; (void)n_in; (void)out_size; (void)ws_size;
    const float* x         = (const float*)d_in[0];
    const float* in_proj_w = (const float*)d_in[1];
    const float* conv_w    = (const float*)d_in[2];
    const float* conv_b    = (const float*)d_in[3];
    const float* x_proj_w  = (const float*)d_in[4];
    const float* dt_w      = (const float*)d_in[5];
    const float* dt_b      = (const float*)d_in[6];
    const float* A_log     = (const float*)d_in[7];
    const float* D_param   = (const float*)d_in[8];
    const float* out_proj_w= (const float*)d_in[9];
    float* out = (float*)d_out;

    float* ws    = (float*)d_ws;
    float* xz    = ws;                                   // 4096 x 4096
    float* xconv = xz    + (size_t)MTOT * 2 * DINNER;    // 4096 x 2048
    float* dbc   = xconv + (size_t)MTOT * DINNER;        // 4096 x 96
    float* delta = dbc   + (size_t)MTOT * (DTRANK + 2 * DSTATE); // 4096 x 2048
    float* ycomb = delta + (size_t)MTOT * DINNER;        // 4096 x 2048

    dim3 blk(256);

    // 1) xz = x @ in_proj_w^T        (M=4096, N=4096, K=1024)
    gemm_tn_wmma<0><<<dim3(4096 / 128, MTOT / 128), blk, 0, stream>>>(
        x, DMODEL, in_proj_w, DMODEL, xz, 2 * DINNER, MTOT, 2 * DINNER, DMODEL, nullptr);

    // 2) depthwise causal conv + SiLU
    conv_silu_kernel<<<(MTOT * DINNER) / 256, blk, 0, stream>>>(xz, conv_w, conv_b, xconv);

    // 3) dbc = xconv @ x_proj_w^T    (M=4096, N=96, K=2048)
    gemm_tn_wmma<0><<<dim3(1, MTOT / 128), blk, 0, stream>>>(
        xconv, DINNER, x_proj_w, DINNER, dbc, DTRANK + 2 * DSTATE,
        MTOT, DTRANK + 2 * DSTATE, DINNER, nullptr);

    // 4) delta = softplus(dbc[:, :64] @ dt_w^T + dt_b)  (M=4096, N=2048, K=64)
    gemm_tn_wmma<1><<<dim3(DINNER / 128, MTOT / 128), blk, 0, stream>>>(
        dbc, DTRANK + 2 * DSTATE, dt_w, DTRANK, delta, DINNER,
        MTOT, DINNER, DTRANK, dt_b);

    // 5) selective scan + D*x + silu(z) gating -> ycomb
    scan_kernel<<<BATCH * (DINNER / 16), blk, 0, stream>>>(
        xconv, dbc, delta, A_log, D_param, xz, ycomb);

    // 6) out = ycomb @ out_proj_w^T  (M=4096, N=1024, K=2048)
    gemm_tn_wmma<0><<<dim3(DMODEL / 128, MTOT / 128), blk, 0, stream>>>(
        ycomb, DINNER, out_proj_w, DINNER, out, DMODEL, MTOT, DMODEL, DINNER, nullptr);
}